// DeepMCGCN_7576322310514
// MI455X (gfx1250) — compile-verified
//
#include <hip/hip_runtime.h>
#include <math.h>

typedef __attribute__((ext_vector_type(2))) float v2f;
typedef __attribute__((ext_vector_type(8))) float v8f;

#define EPSV 1e-5f

// ---------------------------------------------------------------------------
// Node embedding: h[s,b,n,d] = sum_f nf[b,n,f] * Wn[s,f,d]   (f = 8)
// ---------------------------------------------------------------------------
__global__ void k_embed(const float* __restrict__ nf, const float* __restrict__ Wn,
                        float* __restrict__ h) {
  int sbn = blockIdx.x;
  int s = sbn / 3200;
  int bn = sbn % 3200;
  int d = threadIdx.x;
  const float* w = Wn + (size_t)s * 8 * 256;
  const float* x = nf + (size_t)bn * 8;
  float acc = 0.f;
#pragma unroll
  for (int f = 0; f < 8; ++f) acc += x[f] * w[f * 256 + d];
  h[((size_t)s * 3200 + bn) * 256 + d] = acc;
}

// ---------------------------------------------------------------------------
// Fold edge input projections into per-(branch,layer) 4x16 matrices:
//   wfold[s][l][f][c] = sum_d WeX_in[f][d] * We[s][l][d][c]
// ---------------------------------------------------------------------------
__global__ void k_fold(const float* __restrict__ We_in, const float* __restrict__ We1_in,
                       const float* __restrict__ We2_in, const float* __restrict__ We,
                       float* __restrict__ wfold) {
  int t = blockIdx.x * blockDim.x + threadIdx.x;
  if (t >= 576) return;
  int c = t & 15;
  int f = (t >> 4) & 3;
  int l = (t >> 6) % 3;
  int s = t / 192;
  const float* wsl = We + (size_t)((s * 3 + l) * 256) * 16;
  float acc = 0.f;
  if (s == 0) {
    for (int d = 0; d < 256; ++d) acc += We_in[f * 256 + d] * wsl[d * 16 + c];
  } else if (s == 1) {
    if (f < 2)
      for (int d = 0; d < 256; ++d) acc += We1_in[f * 256 + d] * wsl[d * 16 + c];
  } else {
    if (f >= 2)
      for (int d = 0; d < 256; ++d) acc += We2_in[(f - 2) * 256 + d] * wsl[d * 16 + c];
  }
  wfold[t] = acc;
}

// ---------------------------------------------------------------------------
// LayerNorm over last dim (256), optional pre-add (z = LN(y + h)).
// ---------------------------------------------------------------------------
__global__ void k_ln(const float* __restrict__ in, const float* __restrict__ add,
                     const float* __restrict__ g, const float* __restrict__ bta,
                     float* __restrict__ out) {
  __shared__ float red[256];
  int row = blockIdx.x, tid = threadIdx.x;
  float x = in[(size_t)row * 256 + tid];
  if (add) x += add[(size_t)row * 256 + tid];
  red[tid] = x;
  __syncthreads();
  for (int s = 128; s > 0; s >>= 1) {
    if (tid < s) red[tid] += red[tid + s];
    __syncthreads();
  }
  float mu = red[0] * (1.f / 256.f);
  float dx = x - mu;
  __syncthreads();
  red[tid] = dx * dx;
  __syncthreads();
  for (int s = 128; s > 0; s >>= 1) {
    if (tid < s) red[tid] += red[tid + s];
    __syncthreads();
  }
  float var = red[0] * (1.f / 256.f);
  out[(size_t)row * 256 + tid] = dx * rsqrtf(var + EPSV) * g[tid] + bta[tid];
}

// ---------------------------------------------------------------------------
// f32 WMMA GEMM with compile-time K/Nc. One wave per 16x16 tile, 8 waves/block.
// All 8 waves of a block share one 16-row A panel (tilesN % 8 == 0), which is
// staged into LDS via gfx1250 GLOBAL_LOAD_ASYNC_TO_LDS_B128 (ASYNCcnt), then
// consumed as ds_load_b64 with immediate offsets. B is read from global with
// immediate 24-bit instruction offsets (K/Nc compile-time -> no address VALU).
// ---------------------------------------------------------------------------
template <bool RELU, bool ADDRES, int K, int Nc>
__global__ void k_gemm(const float* __restrict__ A, const float* __restrict__ W,
                       const float* __restrict__ RES, float* __restrict__ C, int M) {
  constexpr int WAVES = 8;
  constexpr int tilesN = Nc / 16;
  __shared__ float sA[16 * K];

  const int lane = threadIdx.x & 31;
  const int wave = threadIdx.x >> 5;
  const int tilesTotal = (M >> 4) * tilesN;
  const int tile0 = blockIdx.x * WAVES;
  const int m0 = (tile0 / tilesN) << 4;

  // ---- async-stage contiguous 16xK A panel into LDS (16*K floats) ----
  {
    const float* gp = A + (size_t)m0 * K;
    for (int c = threadIdx.x; c < 4 * K; c += blockDim.x) {  // c = float4 chunk
      const float* g = gp + c * 4;
      unsigned laddr = (unsigned)(size_t)(sA + c * 4);
      asm volatile("global_load_async_to_lds_b128 %0, %1, off"
                   :: "v"(laddr), "v"(g) : "memory");
    }
    asm volatile("s_wait_asynccnt 0x0" ::: "memory");
  }
  __syncthreads();

  const int tile = tile0 + wave;
  if (tile < tilesTotal) {  // wave-uniform; grids are exact so always true
    const int n0 = (tile % tilesN) << 4;
    const int ncol = n0 + (lane & 15);
    const int kh = (lane >> 4) << 1;  // 0 or 2
    const float* Wp = W + (size_t)kh * Nc + ncol;
    const float* sAp = sA + (lane & 15) * K + kh;

    v8f acc = {0.f, 0.f, 0.f, 0.f, 0.f, 0.f, 0.f, 0.f};
    for (int k0 = 0; k0 < K; k0 += 64) {
      if (k0 + 64 < K) __builtin_prefetch(Wp + 64 * Nc, 0, 3);
#pragma unroll
      for (int kk = 0; kk < 64; kk += 4) {
        v2f a = *(const v2f*)(sAp + kk);       // ds_load_b64, imm offset
        v2f b;
        b.x = Wp[kk * Nc];                     // global_load_b32, imm offset
        b.y = Wp[kk * Nc + Nc];
        acc = __builtin_amdgcn_wmma_f32_16x16x4_f32(false, a, false, b, (short)0, acc,
                                                    false, false);
      }
      Wp += 64 * Nc;
      sAp += 64;
    }
#pragma unroll
    for (int r = 0; r < 8; ++r) {
      int row = m0 + r + ((lane >> 4) << 3);
      int col = n0 + (lane & 15);
      float v = acc[r];
      if (ADDRES) v += RES[(size_t)row * Nc + col];
      if (RELU) v = fmaxf(v, 0.f);
      C[(size_t)row * Nc + col] = v;
    }
  }
}

// ---------------------------------------------------------------------------
// Attention, one (batch, head) per block, 8 waves process rows independently.
// Edge mode (WF != null): score += e1(i,j); att = softmax(score) * e2(i,j).
// Per-wave LDS handshakes use gfx1250 s_wait_dscnt (no block sync in the
// divergent row loop).
// ---------------------------------------------------------------------------
__global__ void k_attn(const float* __restrict__ Q, int qStride, int qOff,
                       const float* __restrict__ KV, int kvStride, int kOff, int vOff,
                       const float* __restrict__ EF, const float* __restrict__ WF,
                       float* __restrict__ OUT, int outStride, int outOff) {
  __shared__ float sk[100 * 33];
  __shared__ float sv[100 * 33];
  __shared__ float sq[8 * 32];
  __shared__ float satt[8 * 104];
  const int b = blockIdx.x >> 3;
  const int hh = blockIdx.x & 7;
  const int lane = threadIdx.x & 31;
  const int wave = threadIdx.x >> 5;

  for (int t = threadIdx.x; t < 100 * 32; t += blockDim.x) {
    int j = t >> 5, d = t & 31;
    sk[j * 33 + d] = KV[(size_t)(b * 100 + j) * kvStride + kOff + hh * 32 + d];
    sv[j * 33 + d] = KV[(size_t)(b * 100 + j) * kvStride + vOff + hh * 32 + d];
  }
  __syncthreads();

  float wf1[4], wf2[4];
  if (WF) {
#pragma unroll
    for (int f = 0; f < 4; ++f) {
      wf1[f] = WF[f * 16 + hh];
      wf2[f] = WF[f * 16 + 8 + hh];
    }
  }
  const float scale = 0.17677669529663687f;  // 1/sqrt(32)

  for (int i = wave; i < 100; i += (blockDim.x >> 5)) {
    sq[wave * 32 + lane] = Q[(size_t)(b * 100 + i) * qStride + qOff + hh * 32 + lane];
    asm volatile("s_wait_dscnt 0" ::: "memory");

    float sc[4], e2r[4];
    float mx = -INFINITY;
#pragma unroll
    for (int t = 0; t < 4; ++t) {
      int j = lane + t * 32;
      float sval = -INFINITY, gate = 0.f;
      if (j < 100) {
        float dot = 0.f;
        for (int d = 0; d < 32; ++d) dot += sq[wave * 32 + d] * sk[j * 33 + d];
        sval = dot * scale;
        gate = 1.f;
        if (WF) {
          const float* e = EF + (size_t)((b * 100 + i) * 100 + j) * 4;
          float b1 = 0.f, b2 = 0.f;
#pragma unroll
          for (int f = 0; f < 4; ++f) {
            b1 += wf1[f] * e[f];
            b2 += wf2[f] * e[f];
          }
          sval += b1;
          gate = b2;
        }
      }
      sc[t] = sval;
      e2r[t] = gate;
      mx = fmaxf(mx, sval);
    }
#pragma unroll
    for (int off = 16; off > 0; off >>= 1) mx = fmaxf(mx, __shfl_xor(mx, off));

    float ex[4];
    float lsum = 0.f;
#pragma unroll
    for (int t = 0; t < 4; ++t) {
      int j = lane + t * 32;
      float v = (j < 100) ? __expf(sc[t] - mx) : 0.f;
      ex[t] = v;
      lsum += v;
    }
#pragma unroll
    for (int off = 16; off > 0; off >>= 1) lsum += __shfl_xor(lsum, off);
    float inv = 1.f / lsum;

#pragma unroll
    for (int t = 0; t < 4; ++t) {
      int j = lane + t * 32;
      if (j < 100) satt[wave * 104 + j] = ex[t] * inv * e2r[t];
    }
    asm volatile("s_wait_dscnt 0" ::: "memory");

    float acc = 0.f;
    for (int j = 0; j < 100; ++j) acc += satt[wave * 104 + j] * sv[j * 33 + lane];
    OUT[(size_t)(b * 100 + i) * outStride + outOff + hh * 32 + lane] = acc;
  }
}

// ---------------------------------------------------------------------------
// Branch mixing: h0 += o0+o1+o2 ; h1 += o1+o2 ; h2 += o1+o2
// ---------------------------------------------------------------------------
__global__ void k_combine(float* __restrict__ h, const float* __restrict__ o) {
  size_t i = (size_t)blockIdx.x * blockDim.x + threadIdx.x;
  const size_t off = (size_t)3200 * 256;
  float o0 = o[i], o1 = o[i + off], o2 = o[i + 2 * off];
  float t12 = o1 + o2;
  h[i] += o0 + t12;
  h[i + off] += t12;
  h[i + 2 * off] += t12;
}

// copy h (branch 0) into concat columns [512,768)
__global__ void k_copy_h(const float* __restrict__ h, float* __restrict__ x) {
  int row = blockIdx.x, c = threadIdx.x;
  x[(size_t)row * 768 + 512 + c] = h[(size_t)row * 256 + c];
}

// decoder: out[row] = 10 * tanh((x2[row,:] . Wdec) / 16)
__global__ void k_decode(const float* __restrict__ x, const float* __restrict__ wd,
                         float* __restrict__ out) {
  __shared__ float red[256];
  int row = blockIdx.x, tid = threadIdx.x;
  float p = 0.f;
  for (int c = tid; c < 768; c += 256) p += x[(size_t)row * 768 + c] * wd[c];
  red[tid] = p;
  __syncthreads();
  for (int s = 128; s > 0; s >>= 1) {
    if (tid < s) red[tid] += red[tid + s];
    __syncthreads();
  }
  if (tid == 0) out[row] = 10.f * tanhf(red[0] * (1.f / 16.f));
}

// ---------------------------------------------------------------------------
extern "C" void kernel_launch(void* const* d_in, const int* in_sizes, int n_in,
                              void* d_out, int out_size, void* d_ws, size_t ws_size,
                              hipStream_t stream) {
  (void)in_sizes; (void)n_in; (void)out_size; (void)ws_size;
  const float* nf     = (const float*)d_in[0];
  const float* ef     = (const float*)d_in[1];
  const float* Wn     = (const float*)d_in[2];
  const float* We_in  = (const float*)d_in[3];
  const float* We1_in = (const float*)d_in[4];
  const float* We2_in = (const float*)d_in[5];
  const float* ln1g   = (const float*)d_in[6];
  const float* ln1b   = (const float*)d_in[7];
  const float* Wh     = (const float*)d_in[8];
  const float* We     = (const float*)d_in[9];
  const float* ln2g   = (const float*)d_in[10];
  const float* ln2b   = (const float*)d_in[11];
  const float* W1     = (const float*)d_in[12];
  const float* W2     = (const float*)d_in[13];
  const float* Wm1    = (const float*)d_in[14];
  const float* Wm2    = (const float*)d_in[15];
  const float* Wdec   = (const float*)d_in[16];

  const int R = 3200;  // B*N
  const int D = 256;

  float* ws = (float*)d_ws;
  float* h_stack = ws; ws += (size_t)3 * R * D;
  float* hn      = ws; ws += (size_t)R * D;
  float* qkv     = ws; ws += (size_t)R * 768;
  float* ybuf    = ws; ws += (size_t)R * D;
  float* zbuf    = ws; ws += (size_t)R * D;
  float* mid     = ws; ws += (size_t)R * 1024;
  float* obuf    = ws; ws += (size_t)3 * R * D;
  float* wfold   = ws; ws += 576;
  float* xbuf = qkv;   // (R,768) aliases — layer stage done by then
  float* xmid = mid;   // (R,768) <= (R,1024)
  float* x2   = obuf;  // (R,768) <= (3R,256)

  k_embed<<<3 * R, 256, 0, stream>>>(nf, Wn, h_stack);
  k_fold<<<9, 64, 0, stream>>>(We_in, We1_in, We2_in, We, wfold);

  for (int l = 0; l < 3; ++l) {
    for (int s = 0; s < 3; ++s) {
      const int sl = s * 3 + l;
      k_ln<<<R, 256, 0, stream>>>(h_stack + (size_t)s * R * D, nullptr,
                                  ln1g + (size_t)sl * D, ln1b + (size_t)sl * D, hn);
      k_gemm<false, false, 256, 768><<<(200 * 48) / 8, 256, 0, stream>>>(
          hn, Wh + (size_t)sl * D * 768, nullptr, qkv, R);
      k_attn<<<32 * 8, 256, 0, stream>>>(qkv, 768, 0, qkv, 768, 256, 512, ef,
                                         wfold + (size_t)sl * 64, ybuf, 256, 0);
      k_ln<<<R, 256, 0, stream>>>(ybuf, h_stack + (size_t)s * R * D,
                                  ln2g + (size_t)sl * D, ln2b + (size_t)sl * D, zbuf);
      k_gemm<true, false, 256, 1024><<<(200 * 64) / 8, 256, 0, stream>>>(
          zbuf, W1 + (size_t)sl * D * 1024, nullptr, mid, R);
      k_gemm<false, true, 1024, 256><<<(200 * 16) / 8, 256, 0, stream>>>(
          mid, W2 + (size_t)sl * 1024 * D, ybuf, obuf + (size_t)s * R * D, R);
    }
    k_combine<<<R, 256, 0, stream>>>(h_stack, obuf);
  }

  // final cross-stack attentions: a1 = MHA(q=h2, kv=h1), a2 = MHA(q=h1, kv=h2)
  k_attn<<<256, 256, 0, stream>>>(h_stack + 2 * (size_t)R * D, 256, 0,
                                  h_stack + (size_t)R * D, 256, 0, 0,
                                  nullptr, nullptr, xbuf, 768, 0);
  k_attn<<<256, 256, 0, stream>>>(h_stack + (size_t)R * D, 256, 0,
                                  h_stack + 2 * (size_t)R * D, 256, 0, 0,
                                  nullptr, nullptr, xbuf, 768, 256);
  k_copy_h<<<R, 256, 0, stream>>>(h_stack, xbuf);

  k_gemm<true, false, 768, 768><<<(200 * 48) / 8, 256, 0, stream>>>(xbuf, Wm1, nullptr,
                                                                    xmid, R);
  k_gemm<false, false, 768, 768><<<(200 * 48) / 8, 256, 0, stream>>>(xmid, Wm2, nullptr,
                                                                     x2, R);
  k_decode<<<R, 256, 0, stream>>>(x2, Wdec, (float*)d_out);
}